// GCNLayer_65206193487903
// MI455X (gfx1250) — compile-verified
//
#include <hip/hip_runtime.h>

// GCN layer for MI455X (gfx1250, wave32).
// Pipeline (all buffers L2-resident, scatter payload minimized to 2 floats/edge
// by exploiting linearity of GCN propagation):
//   1) deg      : deg[i] = 1 (self loop) + #incoming edges      (atomic f32)
//   2) dinv     : dinv = rsqrt(deg);  agg1 = dinv^2 * x         (self-loop term)
//   3) agg1     : += norm(r,c) * x[r]  scattered to c           (atomic f32, 2 floats/edge)
//   4) stats    : h = agg1 @ W1 + b1 via V_WMMA_F32_16X16X4_F32; sum/sumsq -> stats[40]
//   5) bn+W2    : recompute h, BatchNorm(train stats)+ReLU, t = . @ W2;
//                 out = dinv^2 * t + b2                          (self-loop term)
//   6) out      : += norm(r,c) * t[r]  scattered to c           (atomic f32)

typedef float v2f __attribute__((ext_vector_type(2)));
typedef float v8f __attribute__((ext_vector_type(8)));

#define HID 20
#define BN_EPS 1e-5f

__global__ void k_init(float* __restrict__ deg, float* __restrict__ stats, int m) {
    int i = blockIdx.x * blockDim.x + threadIdx.x;
    if (i < m) deg[i] = 1.0f;                 // self-loop contribution to degree
    if (i < 64) stats[i] = 0.0f;              // sum[20] | sumsq[20] | pad
}

__global__ void k_deg(const long long* __restrict__ col, float* __restrict__ deg, int e) {
    int i = blockIdx.x * blockDim.x + threadIdx.x;
    if (i < e) unsafeAtomicAdd(&deg[(int)col[i]], 1.0f);
}

__global__ void k_dinv_selfagg(const float* __restrict__ x, float* __restrict__ dinv,
                               float* __restrict__ agg, int m) {
    int i = blockIdx.x * blockDim.x + threadIdx.x;
    if (i >= m) return;
    float d = rsqrtf(dinv[i]);                // dinv[] currently holds deg (>= 1)
    dinv[i] = d;
    float s = d * d;                          // self-loop norm
    agg[2 * i]     = s * x[2 * i];
    agg[2 * i + 1] = s * x[2 * i + 1];
}

// Shared by both propagation layers: dst[c] += dinv[r]*dinv[c] * src[r]  (2-dim features)
__global__ void k_edge_agg2d(const long long* __restrict__ row, const long long* __restrict__ col,
                             const float* __restrict__ dinv, const float* __restrict__ src,
                             float* __restrict__ dst, int e) {
    int i = blockIdx.x * blockDim.x + threadIdx.x;
    if (i >= e) return;
    int r = (int)row[i], c = (int)col[i];
    float nrm = dinv[r] * dinv[c];
    unsafeAtomicAdd(&dst[2 * c],     nrm * src[2 * r]);
    unsafeAtomicAdd(&dst[2 * c + 1], nrm * src[2 * r + 1]);
}

// One wave computes a 16-row x 20-col tile of h = agg1 @ W1 + b1 with two
// V_WMMA_F32_16X16X4_F32 (K padded 2->4, N tiled 16 + 4), then accumulates
// per-channel sum / sum-of-squares through LDS into global stats[40].
__global__ void k_stats_wmma(const float* __restrict__ agg, const float* __restrict__ W1,
                             const float* __restrict__ b1, float* __restrict__ stats, int m) {
    __shared__ float s_red[2 * HID];
    int tid = threadIdx.x;
    if (tid < 2 * HID) s_red[tid] = 0.0f;
    __syncthreads();

    int lane = tid & 31;
    int l    = lane & 15;
    int hi   = lane >> 4;                       // lanes 16-31 hold K=2,3 (zero pad)
    int wave = (blockIdx.x * blockDim.x + tid) >> 5;
    int base = wave * 16;                       // first node row of this tile

    if (base < m) {                             // wave-uniform: EXEC stays all-ones
        // A (16x4 f32): lane<16 -> row l, {K0,K1}; lane>=16 -> {K2,K3} = 0
        float a0 = agg[2 * (base + l)];
        float a1 = agg[2 * (base + l) + 1];
        a0 = hi ? 0.0f : a0;
        a1 = hi ? 0.0f : a1;
        v2f A = {a0, a1};
        // B (4x16 f32): lane<16 -> col l rows {K0,K1}; lane>=16 -> {K2,K3} = 0
        float w00 = W1[l], w01 = W1[HID + l];               // tile0: channels 0..15
        int   c1  = (16 + l < HID) ? (16 + l) : (HID - 1);  // clamp OOB reads
        float w10 = W1[c1], w11 = W1[HID + c1];             // tile1: channels 16..19
        bool ok1 = (!hi) && (l < HID - 16);
        v2f B0 = { hi ? 0.0f : w00, hi ? 0.0f : w01 };
        v2f B1 = { ok1 ? w10 : 0.0f, ok1 ? w11 : 0.0f };
        v8f C  = {0.f, 0.f, 0.f, 0.f, 0.f, 0.f, 0.f, 0.f};
        v8f d0 = __builtin_amdgcn_wmma_f32_16x16x4_f32(false, A, false, B0, (short)0, C, false, false);
        v8f d1 = __builtin_amdgcn_wmma_f32_16x16x4_f32(false, A, false, B1, (short)0, C, false, false);
        // D layout: d[r] = h[base + r + hi*8][channel], channel = l (d0) / 16+l (d1)
        float bias0 = b1[l];
        float s0 = 0.f, q0 = 0.f;
#pragma unroll
        for (int r = 0; r < 8; ++r) { float v = d0[r] + bias0; s0 += v; q0 += v * v; }
        atomicAdd(&s_red[l], s0);
        atomicAdd(&s_red[HID + l], q0);
        if (l < HID - 16) {
            float bias1 = b1[16 + l];
            float s1 = 0.f, q1 = 0.f;
#pragma unroll
            for (int r = 0; r < 8; ++r) { float v = d1[r] + bias1; s1 += v; q1 += v * v; }
            atomicAdd(&s_red[16 + l], s1);
            atomicAdd(&s_red[HID + 16 + l], q1);
        }
    }
    __syncthreads();
    if (tid < 2 * HID) unsafeAtomicAdd(&stats[tid], s_red[tid]);
}

// Recompute h (40 FMAs), BatchNorm (training stats) + ReLU, apply W2 (20->2),
// write t and the self-loop-initialized output.
__global__ void k_bn_w2(const float* __restrict__ agg, const float* __restrict__ dinv,
                        const float* __restrict__ W1, const float* __restrict__ b1,
                        const float* __restrict__ gamma, const float* __restrict__ beta,
                        const float* __restrict__ W2, const float* __restrict__ b2,
                        const float* __restrict__ stats,
                        float* __restrict__ t, float* __restrict__ out, int m) {
    int i = blockIdx.x * blockDim.x + threadIdx.x;
    if (i >= m) return;
    float inv_m = 1.0f / (float)m;
    float a0 = agg[2 * i], a1 = agg[2 * i + 1];
    float acc0 = 0.0f, acc1 = 0.0f;
#pragma unroll
    for (int c = 0; c < HID; ++c) {
        float h    = a0 * W1[c] + a1 * W1[HID + c] + b1[c];
        float mean = stats[c] * inv_m;
        float var  = stats[HID + c] * inv_m - mean * mean;   // biased batch variance
        float v    = (h - mean) * rsqrtf(var + BN_EPS) * gamma[c] + beta[c];
        v = v > 0.0f ? v : 0.0f;                             // ReLU
        acc0 += v * W2[2 * c];
        acc1 += v * W2[2 * c + 1];
    }
    t[2 * i] = acc0;
    t[2 * i + 1] = acc1;
    float d = dinv[i];
    float s = d * d;                                         // self-loop norm
    out[2 * i]     = s * acc0 + b2[0];
    out[2 * i + 1] = s * acc1 + b2[1];
}

extern "C" void kernel_launch(void* const* d_in, const int* in_sizes, int n_in,
                              void* d_out, int out_size, void* d_ws, size_t ws_size,
                              hipStream_t stream) {
    const float*     x     = (const float*)d_in[0];
    const long long* ei    = (const long long*)d_in[1];   // int64 (2, E)
    const float*     W1    = (const float*)d_in[2];       // (2, 20) row-major
    const float*     b1    = (const float*)d_in[3];
    const float*     gamma = (const float*)d_in[4];
    const float*     beta  = (const float*)d_in[5];
    const float*     W2    = (const float*)d_in[6];       // (20, 2) row-major
    const float*     b2    = (const float*)d_in[7];
    float*           out   = (float*)d_out;

    const int m = in_sizes[0] / 2;          // 819200 nodes (multiple of 16)
    const int e = in_sizes[1] / 2;          // 2,000,000 edges
    const long long* row = ei;
    const long long* col = ei + e;

    // Workspace layout (floats): dinv[m] | agg1[2m] | t[2m] | stats[64] ~= 16.4 MB
    float* ws    = (float*)d_ws;
    float* dinv  = ws;                        // deg, then dinv in-place
    float* agg1  = ws + (size_t)m;
    float* tbuf  = ws + 3 * (size_t)m;
    float* stats = ws + 5 * (size_t)m;

    const int B  = 256;                       // 8 wave32 per block
    const int gm = (m + B - 1) / B;
    const int ge = (e + B - 1) / B;
    const int tiles = (m + 15) / 16;
    const int gw = (tiles + 7) / 8;           // one wave per 16-row tile

    k_init        <<<gm, B, 0, stream>>>(dinv, stats, m);
    k_deg         <<<ge, B, 0, stream>>>(col, dinv, e);
    k_dinv_selfagg<<<gm, B, 0, stream>>>(x, dinv, agg1, m);
    k_edge_agg2d  <<<ge, B, 0, stream>>>(row, col, dinv, x, agg1, e);      // layer-1 propagate (2-dim)
    k_stats_wmma  <<<gw, B, 0, stream>>>(agg1, W1, b1, stats, m);          // WMMA GEMM + BN stats
    k_bn_w2       <<<gm, B, 0, stream>>>(agg1, dinv, W1, b1, gamma, beta,
                                         W2, b2, stats, tbuf, out, m);
    k_edge_agg2d  <<<ge, B, 0, stream>>>(row, col, dinv, tbuf, out, e);    // layer-2 propagate (2-dim)
}